// RoIPool_71640054497672
// MI455X (gfx1250) — compile-verified
//
#include <hip/hip_runtime.h>
#include <hip/hip_bf16.h>
#include <stdint.h>

// RoI max-pool (torchvision/mmcv MaxRoiPool semantics) for MI455X / gfx1250.
// x[N=2, C=256, H=50, W=50] fp32, rois[R,5] fp32 -> out[R, C, 7, 7] fp32.
//
// MI455X data path: one 256-thread (8 x wave32) block per (roi, 4-channel
// slab). A single Tensor-Data-Mover descriptor (tensor_load_to_lds, 3D tile:
// fw x fh x 4 channels, row stride W, plane stride H*W) stages the RoI
// footprint into contiguous LDS; wave 0 waits TENSORcnt, workgroup barrier,
// then 4*49 threads compute the bin maxima from LDS.

#define PH 7
#define PW 7
#define SCALE_F 0.0625f
#define C_CH 256
#define H_IN 50
#define W_IN 50
#define CB 4                       // channels per block
#define PLANE (H_IN * W_IN)        // 2500 floats: worst-case footprint
#define NTHREADS 256

typedef uint32_t v4u __attribute__((ext_vector_type(4)));
typedef uint32_t v8u __attribute__((ext_vector_type(8)));

__global__ __launch_bounds__(NTHREADS)
void roi_pool_tdm_kernel(const float* __restrict__ input,
                         const float* __restrict__ rois,
                         float* __restrict__ out) {
    __shared__ float smem[CB * PLANE];   // 40 KB of the 320 KB WGP LDS

    const int cblks = C_CH / CB;
    const int r     = blockIdx.x / cblks;
    const int c0    = (blockIdx.x % cblks) * CB;
    const int tid   = threadIdx.x;

    // ---- uniform RoI decode (rintf == round-half-even == jnp.round) ----
    const float* roi = rois + (size_t)r * 5;
    const int b  = (int)roi[0];
    const int x1 = (int)rintf(roi[1] * SCALE_F);
    const int y1 = (int)rintf(roi[2] * SCALE_F);
    const int x2 = (int)rintf(roi[3] * SCALE_F);
    const int y2 = (int)rintf(roi[4] * SCALE_F);
    const int roiw = max(x2 - x1 + 1, 1);
    const int roih = max(y2 - y1 + 1, 1);
    // Footprint touched by any bin: [x1, min(x1+roiw, W)) x [y1, min(y1+roih, H))
    const int fw = min(x1 + roiw, W_IN) - x1;   // >= 1
    const int fh = min(y1 + roih, H_IN) - y1;   // >= 1
    const int fsize = fw * fh;

    // ---- TDM: one 3D tensor DMA stages CB channel footprints into LDS ----
    {
        const uint64_t gaddr = (uint64_t)(uintptr_t)
            (input + (((size_t)b * C_CH + c0) * H_IN + (size_t)y1) * W_IN + x1);
        // Low 32 bits of a generic LDS pointer are the wave-relative LDS addr.
        const uint32_t ldsaddr = (uint32_t)(uintptr_t)(&smem[0]);

        const uint32_t g_lo  = __builtin_amdgcn_readfirstlane((uint32_t)gaddr);
        const uint32_t g_hi  = __builtin_amdgcn_readfirstlane((uint32_t)(gaddr >> 32));
        const uint32_t s_lds = __builtin_amdgcn_readfirstlane(ldsaddr);
        const uint32_t s_fw  = __builtin_amdgcn_readfirstlane((uint32_t)fw);
        const uint32_t s_fh  = __builtin_amdgcn_readfirstlane((uint32_t)fh);

        v4u g0; v8u g1; v4u g2; v4u g3;
        // --- D# group 0 (128b) ---
        g0[0] = 1u;                                   // count=1 (valid), no gather
        g0[1] = s_lds;                                // lds_addr[31:0]
        g0[2] = g_lo;                                 // global_addr[31:0]
        g0[3] = (g_hi & 0x01FFFFFFu) | (2u << 30);    // global_addr[56:32] | type=2
        // --- D# group 1 (256b) ---
        g1[0] = 2u << 16;                             // wg_mask=0, data_size=4B
        g1[1] = (uint32_t)W_IN << 16;                 // tensor_dim0[15:0] (=50)
        g1[2] = (uint32_t)H_IN << 16;                 // td0[31:16]=0 | tensor_dim1[15:0]
        g1[3] = s_fw << 16;                           // td1[31:16]=0 | tile_dim0=fw
        g1[4] = s_fh | ((uint32_t)CB << 16);          // tile_dim1=fh | tile_dim2=CB
        g1[5] = (uint32_t)W_IN;                       // tensor_dim0_stride[31:0]=W
        g1[6] = (uint32_t)PLANE << 16;                // str0[47:32]=0 | str1[15:0]=H*W
        g1[7] = 0u;                                   // str1[47:16]=0 (2500 < 2^16)
        // --- D# group 2 (128b), iterate disabled ---
        g2[0] = (uint32_t)C_CH;                       // tensor_dim2
        g2[1] = 0u;                                   // tensor_dim3 (tile_dim3 unused)
        g2[2] = 0u;                                   // tensor_dim2_stride[31:0]
        g2[3] = 0u;                                   // str2[47:32]=0 | tile_dim3=0
        // --- D# group 3 (128b) ---
        g3[0] = 0u; g3[1] = 0u; g3[2] = 0u; g3[3] = 0u;

        if (tid < 32) {                               // exactly one wave issues the DMA
            asm volatile("tensor_load_to_lds %0, %1, %2, %3"
                         :: "s"(g0), "s"(g1), "s"(g2), "s"(g3)
                         : "memory");
            __builtin_amdgcn_s_wait_tensorcnt(0);     // drain TENSORcnt
        }
    }
    __syncthreads();                                  // LDS tile visible to all waves

    // ---- compute 7x7 bin maxima for CB channels out of LDS ----
    if (tid < CB * (PH * PW)) {
        const int cl  = tid / (PH * PW);
        const int bin = tid - cl * (PH * PW);
        const int ph  = bin / PW;
        const int pw  = bin - ph * PW;

        const float bw = (float)roiw / (float)PW;
        const float bh = (float)roih / (float)PH;
        int ws = min(max((int)floorf((float)pw * bw)         + x1, 0), W_IN);
        int we = min(max((int)ceilf(((float)pw + 1.0f) * bw) + x1, 0), W_IN);
        int hs = min(max((int)floorf((float)ph * bh)         + y1, 0), H_IN);
        int he = min(max((int)ceilf(((float)ph + 1.0f) * bh) + y1, 0), H_IN);

        float m = 0.0f;                        // empty bin -> 0 (reference)
        if ((we > ws) && (he > hs)) {
            m = -__builtin_inff();
            const float* pl = &smem[cl * fsize];   // TDM packs planes at fh*fw
            for (int h = hs; h < he; ++h) {
                const float* rowp = pl + (size_t)(h - y1) * fw - x1;
                for (int w = ws; w < we; ++w) m = fmaxf(m, rowp[w]);
            }
        }
        out[((((size_t)r * C_CH) + (c0 + cl)) * PH + ph) * PW + pw] = m;
    }
}

extern "C" void kernel_launch(void* const* d_in, const int* in_sizes, int n_in,
                              void* d_out, int out_size, void* d_ws, size_t ws_size,
                              hipStream_t stream) {
    const float* input = (const float*)d_in[0];   // [N, C, H, W] fp32
    const float* rois  = (const float*)d_in[1];   // [R, 5] fp32
    float* out = (float*)d_out;                   // [R, C, 7, 7] fp32

    const int R = in_sizes[1] / 5;
    const int cblks = C_CH / CB;
    dim3 grid((unsigned)(R * cblks));
    roi_pool_tdm_kernel<<<grid, NTHREADS, 0, stream>>>(input, rois, out);
}